// ModuleDSepconv_44547400794803
// MI455X (gfx1250) — compile-verified
//
#include <hip/hip_runtime.h>
#include <cstdint>
#include <cstddef>

// ---------------------------------------------------------------------------
// Deformable separable convolution (DSepConv) for MI455X / gfx1250.
// B=4, C=3, H=W=384, F=5 (25 taps). One thread per output pixel, computing
// all 3 channels (shares the 25 bilinear address computations across C).
//
// CDNA5-specific path: the streaming offx/offy/mask reads (the 177 MB that
// dominates HBM traffic) are staged through LDS with the async global->LDS
// engine (ASYNCcnt), double-buffered per tap. Bilinear gathers from the 7 MB
// input plane are direct global loads (cache-resident in WGP$/L2).
// ---------------------------------------------------------------------------

#define TPB 256

typedef int __attribute__((address_space(1)))* gas_iptr;
typedef int __attribute__((address_space(3)))* las_iptr;

__device__ __forceinline__ void async_copy_f32(const float* g, float* l) {
#if __has_builtin(__builtin_amdgcn_global_load_async_to_lds_b32)
  __builtin_amdgcn_global_load_async_to_lds_b32((gas_iptr)g, (las_iptr)l, 0, 0);
#else
  // VDST = LDS byte address (low 32 bits of the generic LDS pointer),
  // VADDR = 64-bit global address, no SADDR.
  unsigned lds_off = (unsigned)(uintptr_t)l;
  asm volatile("global_load_async_to_lds_b32 %0, %1, off"
               :: "v"(lds_off), "v"(g)
               : "memory");
#endif
}

__device__ __forceinline__ void wait_async_le3() {
  asm volatile("s_wait_asynccnt 3" ::: "memory");
}
__device__ __forceinline__ void wait_async_le0() {
  asm volatile("s_wait_asynccnt 0" ::: "memory");
}
__device__ __forceinline__ void wait_ds0() {
  asm volatile("s_wait_dscnt 0" ::: "memory");
}

__global__ __launch_bounds__(TPB)
void dsepconv_kernel(const float* __restrict__ inp,
                     const float* __restrict__ vert,
                     const float* __restrict__ horiz,
                     const float* __restrict__ offx,
                     const float* __restrict__ offy,
                     const float* __restrict__ mask,
                     float* __restrict__ out) {
  constexpr int B  = 4, C = 3, H = 384, W = 384, F = 5;
  constexpr int FF = F * F;
  constexpr int HW = H * W;
  (void)B;

  __shared__ float s_ox[2][TPB];
  __shared__ float s_oy[2][TPB];
  __shared__ float s_m [2][TPB];

  const int tid = threadIdx.x;
  const int p   = blockIdx.x * TPB + tid;   // grid sized exactly: B*HW % TPB == 0
  const int b   = p / HW;
  const int pix = p - b * HW;
  const int h   = pix / W;
  const int w   = pix - h * W;

  const float* obx = offx + (size_t)b * FF * HW + pix;   // k-plane stride = HW
  const float* oby = offy + (size_t)b * FF * HW + pix;
  const float* obm = mask + (size_t)b * FF * HW + pix;
  const float* vb  = vert  + (size_t)b * F * HW + pix;
  const float* hb  = horiz + (size_t)b * F * HW + pix;
  const float* inb = inp + (size_t)b * C * HW;

  // Separable tap weights for this pixel (coalesced plane reads).
  float vh[F], hz[F];
#pragma unroll
  for (int i = 0; i < F; ++i) {
    vh[i] = vb[(size_t)i * HW];
    hz[i] = hb[(size_t)i * HW];
  }

  // Prologue: stage tap k=0 (offx, offy, mask) into LDS buffer 0.
  async_copy_f32(obx, &s_ox[0][tid]);
  async_copy_f32(oby, &s_oy[0][tid]);
  async_copy_f32(obm, &s_m [0][tid]);

  float acc[C];
#pragma unroll
  for (int c = 0; c < C; ++c) acc[c] = 0.0f;

#pragma unroll
  for (int ki = 0; ki < F; ++ki) {
#pragma unroll
    for (int kj = 0; kj < F; ++kj) {
      const int k   = ki * F + kj;
      const int cur = k & 1;

      if (k + 1 < FF) {
        // Prefetch tap k+1 into the other buffer, then drain tap k's copies.
        const int    nxt = cur ^ 1;
        const size_t o   = (size_t)(k + 1) * HW;
        async_copy_f32(obx + o, &s_ox[nxt][tid]);
        async_copy_f32(oby + o, &s_oy[nxt][tid]);
        async_copy_f32(obm + o, &s_m [nxt][tid]);
        wait_async_le3();   // in-order completion => tap k's 3 copies landed
      } else {
        wait_async_le0();
      }

      float ox = s_ox[cur][tid];
      float oy = s_oy[cur][tid];
      float m  = s_m [cur][tid];
      // DS reads must complete before the NEXT iteration overwrites `cur`.
      wait_ds0();

      // Deformed sample position, clipped exactly like the reference.
      float py = oy + (float)h + ((float)ki - 2.0f);
      float px = ox + (float)w + ((float)kj - 2.0f);
      py = fminf(fmaxf(py, 0.0f), (float)(H - 1));
      px = fminf(fmaxf(px, 0.0f), (float)(W - 1));

      const float y0f = floorf(py);
      const float x0f = floorf(px);
      const float wy  = py - y0f;
      const float wx  = px - x0f;
      int y0 = (int)y0f;
      int x0 = (int)x0f;
      int y1 = y0 + 1; if (y1 > H - 1) y1 = H - 1;
      int x1 = x0 + 1; if (x1 > W - 1) x1 = W - 1;

      const float omwy = 1.0f - wy;
      const float omwx = 1.0f - wx;
      const float w00  = omwy * omwx;
      const float w01  = omwy * wx;
      const float w10  = wy * omwx;
      const float w11  = wy * wx;

      const float wt = vh[ki] * hz[kj] * m;

      const int i00 = y0 * W + x0;
      const int i01 = y0 * W + x1;
      const int i10 = y1 * W + x0;
      const int i11 = y1 * W + x1;

#pragma unroll
      for (int c = 0; c < C; ++c) {
        const float* pc = inb + (size_t)c * HW;
        float s = fmaf(pc[i00], w00,
                  fmaf(pc[i01], w01,
                  fmaf(pc[i10], w10, pc[i11] * w11)));
        acc[c] = fmaf(wt, s, acc[c]);
      }
    }
  }

  const size_t ob = (size_t)b * C * HW + pix;
#pragma unroll
  for (int c = 0; c < C; ++c) out[ob + (size_t)c * HW] = acc[c];
}

// ---------------------------------------------------------------------------

extern "C" void kernel_launch(void* const* d_in, const int* in_sizes, int n_in,
                              void* d_out, int out_size, void* d_ws, size_t ws_size,
                              hipStream_t stream) {
  (void)in_sizes; (void)n_in; (void)out_size; (void)d_ws; (void)ws_size;

  const float* inp   = (const float*)d_in[0];
  const float* vert  = (const float*)d_in[1];
  const float* horiz = (const float*)d_in[2];
  const float* offx  = (const float*)d_in[3];
  const float* offy  = (const float*)d_in[4];
  const float* mask  = (const float*)d_in[5];
  float*       out   = (float*)d_out;

  constexpr int B = 4, H = 384, W = 384;
  const int N = B * H * W;              // 589824, exactly divisible by TPB
  dim3 grid((N + TPB - 1) / TPB);
  dim3 block(TPB);
  dsepconv_kernel<<<grid, block, 0, stream>>>(inp, vert, horiz, offx, offy, mask, out);
}